// HoltWintersDecomposition_75539884802578
// MI455X (gfx1250) — compile-verified
//
#include <hip/hip_runtime.h>
#include <stdint.h>

// Holt-Winters decomposition: x (R=8192 rows, T=2048 steps) -> level, season, y.
// Sequential nonlinear scan per row; parallel across rows only (256 wave32s).
// Strategy: 1 thread per row, TDM double-buffered tile staging into LDS,
// ds_load_b128 tile reads, rcp+NR divisions, vectorized NT stores.

#define T_LEN 2048
#define TT 64                       // timesteps per tile
#define ROWS 32                     // rows per block == one wave32
#define ROW_STRIDE_DW 68            // 64 data dwords + 4 pad dwords (TDM pad) -> 16B-aligned rows
#define TILE_DW (ROWS * ROW_STRIDE_DW) // 2176 dwords per buffer
#define NTILES (T_LEN / TT)         // 32
#define EPSV 1e-8f

typedef uint32_t u32;
typedef __attribute__((ext_vector_type(4))) u32   v4u;
typedef __attribute__((ext_vector_type(8))) int   v8i;
typedef __attribute__((ext_vector_type(4))) int   v4i;
typedef __attribute__((ext_vector_type(4))) float v4f;

__device__ __forceinline__ float fast_rcp(float d) {
  float r = __builtin_amdgcn_rcpf(d);     // v_rcp_f32
  r = r * (2.0f - d * r);                 // one Newton-Raphson step -> ~2 ulp
  return r;
}

// Issue a TDM load of a [ROWS x TT] f32 tile (row stride T_LEN elements in
// global) into LDS at byte offset lds_off, with 4-dword padding after every
// 64 dwords (LDS row stride = 68 dwords, keeps 16B alignment for b128 reads).
__device__ __forceinline__ void tdm_load_tile(const float* g, u32 lds_off, u32 nrows) {
  u32 lo = (u32)(uintptr_t)g;
  u32 hi = (u32)(((uintptr_t)g) >> 32);
  v4u g0;
  g0[0] = 1u;                                      // count=1, user descriptor
  g0[1] = lds_off;                                 // LDS byte address
  g0[2] = lo;                                      // global_addr[31:0]
  g0[3] = (hi & 0x01FFFFFFu) | 0x80000000u;        // global_addr[56:32] | type=2
  v8i g1;
  g1[0] = (2 << 16)        // data_size = 4 bytes
        | (1 << 20)        // pad_enable
        | (5 << 22)        // pad_interval: 64 dwords
        | (3 << 25);       // pad_amount: 4 dwords
  g1[1] = (int)(((u32)(T_LEN & 0xFFFF)) << 16);              // tensor_dim0[15:0] in bits 63:48
  g1[2] = (int)(((u32)T_LEN >> 16) | ((nrows & 0xFFFFu) << 16)); // dim0 hi | tensor_dim1 lo
  g1[3] = (int)((nrows >> 16) | ((u32)TT << 16));            // dim1 hi | tile_dim0 = 64
  g1[4] = (int)ROWS;                                         // tile_dim1 = 32 (tile_dim2 = 0)
  g1[5] = (int)T_LEN;                                        // tensor_dim0_stride[31:0] = 2048
  g1[6] = 0;                                                 // stride hi | dim1_stride lo
  g1[7] = 0;
  v4i z4 = {0, 0, 0, 0};
  v8i z8 = {0, 0, 0, 0, 0, 0, 0, 0};
  // clang-23 / amdgpu-toolchain form: 6 args (g0, g1, g2, g3, g_extra, cpol)
  __builtin_amdgcn_tensor_load_to_lds(g0, g1, z4, z4, z8, 0);
}

__global__ __launch_bounds__(ROWS)
void HoltWinters_tdm_scan_kernel(const float* __restrict__ x,
                                 const float* __restrict__ plog_a,
                                 const float* __restrict__ plog_g,
                                 float* __restrict__ out, int R) {
  __shared__ __attribute__((aligned(16))) float lds[2 * TILE_DW];

  const int lane = threadIdx.x;                 // 0..31, one row per lane
  const int row  = blockIdx.x * ROWS + lane;

  const float la = plog_a[0];
  const float lg = plog_g[0];
  const float alpha = 1.0f / (1.0f + __expf(-la));
  const float gamma = 1.0f / (1.0f + __expf(-lg));
  const float oma = 1.0f - alpha;
  const float omg = 1.0f - gamma;

  float* __restrict__ level  = out;
  float* __restrict__ season = out + (size_t)R * T_LEN;
  float* __restrict__ yout   = out + 2ull * (size_t)R * T_LEN;

  const float* rowbase = x + (size_t)blockIdx.x * ROWS * T_LEN;
  // LDS flat addr low 32 bits are the LDS byte offset (aperture tag is in the
  // high dword), so this yields the byte offset the TDM descriptor wants.
  const u32 lds_base = (u32)(uintptr_t)(&lds[0]);

  // Prime the double buffer: tiles 0 and 1 in flight.
  tdm_load_tile(rowbase + 0 * TT, lds_base + 0u,                (u32)R);
  tdm_load_tile(rowbase + 1 * TT, lds_base + (u32)TILE_DW * 4u, (u32)R);

  float l = 0.0f, s = 1.0f;

  for (int tile = 0; tile < NTILES; ++tile) {
    // Tensor loads complete in order: allow 1 outstanding (the prefetched
    // next tile), except on the final tile where nothing follows.
    if (tile == NTILES - 1)
      __builtin_amdgcn_s_wait_tensorcnt(0);
    else
      __builtin_amdgcn_s_wait_tensorcnt(1);
    asm volatile("" ::: "memory");   // TDM wrote LDS: no load may hoist above

    // Opaque per-tile LDS row index: keeps the loads un-foldable while the
    // shared array stays in addrspace(3) -> ds_load_b128.
    int rb = (tile & 1) * TILE_DW + lane * ROW_STRIDE_DW;
    asm volatile("" : "+v"(rb));
    const int tbase = tile * TT;

    for (int x4 = 0; x4 < TT; x4 += 4) {
      const v4f xq = *(const v4f*)&lds[rb + x4];   // ds_load_b128, 16B aligned
      float lv[4], sv[4], yv[4];
#pragma unroll
      for (int j = 0; j < 4; ++j) {
        const float xv = xq[j];
        if (tbase + x4 + j == 0) {        // t == 0 init (uniform branch, tile 0 only)
          l = xv;
          s = 1.0f;
        } else {
          const float rs = fast_rcp(s + EPSV);
          l = fmaf(alpha, xv * rs, oma * l);
          const float rl = fast_rcp(l + EPSV);
          s = fmaf(gamma, xv * rl, omg * s);
        }
        lv[j] = l;
        sv[j] = s;
        yv[j] = xv * fast_rcp(fmaf(l, s, EPSV));
      }
      const size_t o = (size_t)row * T_LEN + (size_t)(tbase + x4);
      v4f L = {lv[0], lv[1], lv[2], lv[3]};
      v4f S = {sv[0], sv[1], sv[2], sv[3]};
      v4f Y = {yv[0], yv[1], yv[2], yv[3]};
      __builtin_nontemporal_store(L, (v4f*)(level  + o));
      __builtin_nontemporal_store(S, (v4f*)(season + o));
      __builtin_nontemporal_store(Y, (v4f*)(yout   + o));
    }

    // Ensure all LDS reads of this buffer retired before TDM overwrites it.
    asm volatile("s_wait_dscnt 0" ::: "memory");
    if (tile + 2 < NTILES) {
      tdm_load_tile(rowbase + (tile + 2) * TT,
                    lds_base + (u32)((tile & 1) * TILE_DW) * 4u, (u32)R);
    }
  }
}

extern "C" void kernel_launch(void* const* d_in, const int* in_sizes, int n_in,
                              void* d_out, int out_size, void* d_ws, size_t ws_size,
                              hipStream_t stream) {
  const float* x  = (const float*)d_in[0];
  const float* la = (const float*)d_in[1];
  const float* lg = (const float*)d_in[2];
  float* out = (float*)d_out;

  const int total = in_sizes[0];        // R * T_LEN
  const int R = total / T_LEN;          // 8192

  dim3 grid(R / ROWS);                  // 256 single-wave workgroups
  dim3 block(ROWS);                     // 32 threads = 1 wave32
  HoltWinters_tdm_scan_kernel<<<grid, block, 0, stream>>>(x, la, lg, out, R);
}